// DMoN_67723044323357
// MI455X (gfx1250) — compile-verified
//
#include <hip/hip_runtime.h>
#include <hip/hip_bf16.h>

// GCNConv + softmax pooling for MI455X (gfx1250, wave32).
//
// Pipeline (all L2-resident intermediates; HBM traffic ~110MB -> ~5us @ 23.3TB/s):
//   1) deg[i] = 1 (self loop) + scatter-count over dst      (atomic f32 add)
//   2) dinv[i] = rsqrt(deg[i])
//   3) h = x @ W1 via V_WMMA_F32_16X16X4_F32 (exact fp32; 16x16 tile / wave,
//      K=128 -> 32 WMMAs per tile)
//   4) agg[i] = h[i]*dinv[i]^2 (self-loop term)
//   5) per (edge,feature) scatter: agg[dst,f] += h[src,f]*dinv[src]*dinv[dst]
//      (16 lanes per edge -> coalesced 64B gather + 64B atomic burst, L2-bound)
//   6) per node: relu(agg+b1) @ Wp + bp, softmax over 16 classes -> out

typedef __attribute__((ext_vector_type(2))) float v2f;
typedef __attribute__((ext_vector_type(8))) float v8f;

#define F_IN 128
#define HID  16
#define NCLS 16

// ---------------------------------------------------------------- degree init
__global__ void gcn_init_deg(float* __restrict__ deg, int n) {
  int i = blockIdx.x * blockDim.x + threadIdx.x;
  if (i < n) deg[i] = 1.0f;  // self loop contributes 1 to every node
}

__global__ void gcn_deg_accum(const int* __restrict__ ei, float* __restrict__ deg, int E) {
  int e = blockIdx.x * blockDim.x + threadIdx.x;
  if (e < E) unsafeAtomicAdd(&deg[ei[E + e]], 1.0f);  // dst row of edge_index
}

__global__ void gcn_dinv(float* __restrict__ deg, int n) {
  int i = blockIdx.x * blockDim.x + threadIdx.x;
  if (i < n) {
    float d = deg[i];
    deg[i] = d > 0.0f ? rsqrtf(d) : 0.0f;  // deg >= 1 always, but match reference
  }
}

// ------------------------------------------------- h = x @ W1 via f32 WMMA
// One wave computes a 16(rows) x 16(HID) output tile. K swept 128 in steps of 4
// with V_WMMA_F32_16X16X4_F32 (exact fp32 accumulate).
//
// A 16x4 f32 layout: lane l<16 -> row M=l, holds K=k+0 (v.x), K=k+1 (v.y);
//                    lane l+16 -> row M=l, holds K=k+2 (v.x), K=k+3 (v.y).
// B 4x16 f32 layout: lane l<16 -> N=l, holds K=k+0 (v.x), K=k+1 (v.y);
//                    lane l+16 -> N=l, holds K=k+2 (v.x), K=k+3 (v.y).
// D 16x16 f32: VGPR j -> M=j (lanes 0-15) / M=j+8 (lanes 16-31), N = lane&15.
__global__ __launch_bounds__(256) void gcn_gemm_wmma(const float* __restrict__ x,
                                                     const float* __restrict__ W,
                                                     float* __restrict__ h,
                                                     int nTiles, int N) {
  const int lane = threadIdx.x & 31;
  const int tile = blockIdx.x * (blockDim.x >> 5) + (threadIdx.x >> 5);
  if (tile >= nTiles) return;  // wave-uniform: EXEC stays all-ones for WMMA

  const int half = lane >> 4;   // 0: K pair {0,1}; 1: K pair {2,3}
  const int l    = lane & 15;
  const int row  = tile * 16 + l;
  const int rowc = row < N ? row : N - 1;  // clamp loads; stores predicated below

  const float* xr = x + (size_t)rowc * F_IN + half * 2;

  v8f acc = {0.f, 0.f, 0.f, 0.f, 0.f, 0.f, 0.f, 0.f};

#pragma unroll
  for (int k = 0; k < F_IN; k += 4) {
    v2f a;
    a.x = xr[k];
    a.y = xr[k + 1];
    const float* wp = W + (size_t)(k + half * 2) * HID + l;
    v2f b;
    b.x = wp[0];
    b.y = wp[HID];
    // (neg_a, A, neg_b, B, c_mod, C, reuse_a, reuse_b)
    acc = __builtin_amdgcn_wmma_f32_16x16x4_f32(false, a, false, b,
                                                (short)0, acc, false, false);
  }

  const int mbase = tile * 16 + half * 8;
#pragma unroll
  for (int j = 0; j < 8; ++j) {
    int m = mbase + j;
    if (m < N) h[(size_t)m * HID + l] = acc[j];
  }
}

// ------------------------------------------ agg init with self-loop message
__global__ void gcn_init_agg(const float* __restrict__ h, const float* __restrict__ dinv,
                             float* __restrict__ agg, int n) {
  int t = blockIdx.x * blockDim.x + threadIdx.x;
  if (t < n * HID) {
    float di = dinv[t >> 4];
    agg[t] = h[t] * di * di;  // self-loop norm = dinv[i]*dinv[i]
  }
}

// -------------------------------------- edge gather/scale/scatter (L2-bound)
__global__ void gcn_edge_scatter(const int* __restrict__ ei, const float* __restrict__ h,
                                 const float* __restrict__ dinv, float* __restrict__ agg,
                                 int E) {
  long long t = (long long)blockIdx.x * blockDim.x + threadIdx.x;
  int e = (int)(t >> 4);
  if (e >= E) return;
  int f = (int)(t & 15);
  int s = ei[e];       // src
  int d = ei[E + e];   // dst
  float val = h[(size_t)s * HID + f] * (dinv[s] * dinv[d]);
  unsafeAtomicAdd(&agg[(size_t)d * HID + f], val);
}

// -------------------- bias + relu + pooling linear + softmax, one node/thread
__global__ void gcn_final(const float* __restrict__ agg, const float* __restrict__ b1,
                          const float* __restrict__ Wp, const float* __restrict__ bp,
                          float* __restrict__ out, int n) {
  int i = blockIdx.x * blockDim.x + threadIdx.x;
  if (i >= n) return;

  float hb[HID];
#pragma unroll
  for (int k = 0; k < HID; ++k) {
    float v = agg[(size_t)i * HID + k] + b1[k];
    hb[k] = v > 0.0f ? v : 0.0f;
  }

  float logits[NCLS];
  float mx = -3.0e38f;
#pragma unroll
  for (int c = 0; c < NCLS; ++c) {
    float accv = bp[c];
#pragma unroll
    for (int k = 0; k < HID; ++k) accv += hb[k] * Wp[k * NCLS + c];  // scalar (uniform) loads of Wp
    logits[c] = accv;
    mx = fmaxf(mx, accv);
  }

  float sum = 0.0f;
#pragma unroll
  for (int c = 0; c < NCLS; ++c) {
    float ev = __expf(logits[c] - mx);
    logits[c] = ev;
    sum += ev;
  }
  float inv = 1.0f / sum;
#pragma unroll
  for (int c = 0; c < NCLS; ++c) out[(size_t)i * NCLS + c] = logits[c] * inv;
}

// ---------------------------------------------------------------------------
extern "C" void kernel_launch(void* const* d_in, const int* in_sizes, int n_in,
                              void* d_out, int out_size, void* d_ws, size_t ws_size,
                              hipStream_t stream) {
  const float* x  = (const float*)d_in[0];
  const float* W1 = (const float*)d_in[1];
  const float* b1 = (const float*)d_in[2];
  const float* Wp = (const float*)d_in[3];
  const float* bp = (const float*)d_in[4];
  const int*   ei = (const int*)d_in[5];

  const int N = in_sizes[0] / F_IN;   // 100000
  const int E = in_sizes[5] / 2;      // 3.2M
  float* out = (float*)d_out;

  // workspace layout: deg/dinv [N] | h [N*HID] | agg [N*HID]  (= 33N floats)
  float* deg = (float*)d_ws;
  float* h   = deg + N;
  float* agg = h + (size_t)N * HID;

  const int tpb = 256;

  gcn_init_deg<<<(N + tpb - 1) / tpb, tpb, 0, stream>>>(deg, N);
  gcn_deg_accum<<<(E + tpb - 1) / tpb, tpb, 0, stream>>>(ei, deg, E);
  gcn_dinv<<<(N + tpb - 1) / tpb, tpb, 0, stream>>>(deg, N);

  const int nTiles = (N + 15) / 16;                       // 6250 wave-tiles
  gcn_gemm_wmma<<<(nTiles + 7) / 8, 256, 0, stream>>>(x, W1, h, nTiles, N);

  const long long nh = (long long)N * HID;
  gcn_init_agg<<<(int)((nh + tpb - 1) / tpb), tpb, 0, stream>>>(h, deg, agg, N);

  const long long tot = (long long)E * HID;               // 51.2M lanes
  gcn_edge_scatter<<<(int)((tot + tpb - 1) / tpb), tpb, 0, stream>>>(ei, h, deg, agg, E);

  gcn_final<<<(N + tpb - 1) / tpb, tpb, 0, stream>>>(agg, b1, Wp, bp, out, N);
}